// TriplaneSampler_56307021250767
// MI455X (gfx1250) — compile-verified
//
#include <hip/hip_runtime.h>
#include <hip/hip_bf16.h>
#include <math.h>

typedef __attribute__((ext_vector_type(2)))  float    v2f;
typedef __attribute__((ext_vector_type(4)))  float    v4f;
typedef __attribute__((ext_vector_type(8)))  float    v8f;
typedef __attribute__((ext_vector_type(16))) _Float16 v16h;

#define NB      4
#define CTOT    56
#define CFEAT   32
#define NPARTS  24
#define HH      256
#define NPTS    8192

// ---------------------------------------------------------------------------
// Pass 1: meanfeat[b,k,y,x] = (1/32) * sum_{c<32} triplane[b,c,y,k*256+x]
// One wave32 handles 16 consecutive x-pixels of one (b,k,y) row.
// Channel reduction via V_WMMA_F32_16X16X4_F32: A = 16 pixels x 4 channels,
// B = broadcast 1/32 (all elements equal -> B lane layout irrelevant),
// 8 accumulating WMMAs sum all 32 feature channels in full f32 precision.
// ---------------------------------------------------------------------------
__global__ __launch_bounds__(256) void mean_kernel(const float* __restrict__ tp,
                                                   float* __restrict__ mf) {
    const int lane  = threadIdx.x & 31;
    const int gwave = blockIdx.x * 8 + (threadIdx.x >> 5);
    const int pixbase = gwave * 16;          // linear pixel over (b,k,y,x), x fastest
    const int x0 = pixbase & 255;
    const int y  = (pixbase >> 8) & 255;
    const int bk = pixbase >> 16;
    const int b  = bk / 3, k = bk % 3;
    const int L    = lane & 15;
    const int half = lane >> 4;
    const int x    = x0 + L;

    // &triplane[b, c=0, y, k*256 + x]; channel stride = 256*768 floats
    const float* src = tp + ((size_t)(b * CTOT) * HH + y) * (3 * HH) + k * HH + x;
    const size_t cstride = (size_t)HH * (3 * HH);

    v8f acc = {0.f, 0.f, 0.f, 0.f, 0.f, 0.f, 0.f, 0.f};

    // A (16x4 f32): lanes 0-15 -> K=0 (v0), K=1 (v1); lanes 16-31 -> K=2,3.
    const v2f bscale = {1.0f / 32.0f, 1.0f / 32.0f};
    #pragma unroll
    for (int c0 = 0; c0 < CFEAT; c0 += 4) {
        const int ca = c0 + half * 2;
        v2f a;
        a.x = src[(size_t)ca * cstride];
        a.y = src[(size_t)(ca + 1) * cstride];
        acc = __builtin_amdgcn_wmma_f32_16x16x4_f32(
            false, a, false, bscale, (short)0, acc, false, false);
    }

    // D layout (32-bit C/D 16x16): VGPR r -> lanes 0-15 = row r, lanes 16-31
    // = row 8+r; all 16 columns identical (B constant). So lane 0 holds pixels
    // 0..7 in acc[0..7], lane 16 holds pixels 8..15. Vectorized b128 stores.
    float* dst = mf + (size_t)pixbase;       // 64-byte aligned (pixbase % 16 == 0)
    const v4f lo = {acc[0], acc[1], acc[2], acc[3]};
    const v4f hi = {acc[4], acc[5], acc[6], acc[7]};
    if (lane == 0) {
        *(v4f*)(dst + 0) = lo;
        *(v4f*)(dst + 4) = hi;
    }
    if (lane == 16) {
        *(v4f*)(dst + 8)  = lo;
        *(v4f*)(dst + 12) = hi;
    }
}

// ---------------------------------------------------------------------------
// Pass 2: one wave32 per query point; lane p (<24) handles part p.
// Per part: 3 planes x 4 corners; gather meanfeat (3 MB, L2/L0-resident) and
// one logit channel (75 MB region, L2-resident) per corner -> 2 floats/corner
// instead of 33 channels (mean commutes with the bilinear weights).
// Then wave32 softmax over 24 parts and weighted sum -> 1 float per point.
// ---------------------------------------------------------------------------
__global__ __launch_bounds__(256) void sample_kernel(const float* __restrict__ tp,
                                                     const float* __restrict__ coord,
                                                     const int* __restrict__ index,
                                                     const float* __restrict__ mf,
                                                     float* __restrict__ out) {
    const int lane  = threadIdx.x & 31;
    const int point = blockIdx.x * 8 + (threadIdx.x >> 5);   // 0 .. 32767
    const int b     = point >> 13;                           // / NPTS
    const int idx   = index[point];
    const int p     = (lane < NPARTS) ? lane : (NPARTS - 1);

    const size_t row = (size_t)idx * NPARTS + p;
    const float cx = coord[row * 3 + 0];
    const float cy = coord[row * 3 + 1];
    const float cz = coord[row * 3 + 2];

    const float* lbase = tp + (size_t)(b * CTOT + CFEAT + p) * HH * (3 * HH);
    const float* mbase = mf + (size_t)(b * 3) * HH * HH;

    float feat = 0.f, logit = 0.f;
    #pragma unroll
    for (int k = 0; k < 3; ++k) {
        const float u = (k == 2) ? cy : cx;          // k0:(x,y) k1:(x,z) k2:(y,z)
        const float v = (k == 0) ? cy : cz;
        const float px = fmaf(u, 128.f, 127.5f);     // (u+1)*128 - 0.5
        const float py = fmaf(v, 128.f, 127.5f);
        const float xf = floorf(px), yf = floorf(py);
        const float wx = px - xf,    wy = py - yf;
        const int ix0 = min(max((int)xf, 0), HH - 1);
        const int ix1 = min(max((int)xf + 1, 0), HH - 1);
        const int iy0 = min(max((int)yf, 0), HH - 1);
        const int iy1 = min(max((int)yf + 1, 0), HH - 1);
        const float w00 = (1.f - wx) * (1.f - wy);
        const float w01 = wx * (1.f - wy);
        const float w10 = (1.f - wx) * wy;
        const float w11 = wx * wy;

        const float* m = mbase + (size_t)k * HH * HH;
        feat += w00 * m[iy0 * HH + ix0] + w01 * m[iy0 * HH + ix1]
              + w10 * m[iy1 * HH + ix0] + w11 * m[iy1 * HH + ix1];

        const float* l = lbase + k * HH;
        logit += w00 * l[(size_t)iy0 * (3 * HH) + ix0]
               + w01 * l[(size_t)iy0 * (3 * HH) + ix1]
               + w10 * l[(size_t)iy1 * (3 * HH) + ix0]
               + w11 * l[(size_t)iy1 * (3 * HH) + ix1];
    }

    if (lane >= NPARTS) logit = -INFINITY;

    // wave32 reductions
    float mx = logit;
    #pragma unroll
    for (int off = 16; off > 0; off >>= 1)
        mx = fmaxf(mx, __shfl_xor(mx, off, 32));
    const float e  = (lane < NPARTS) ? expf(logit - mx) : 0.f;
    float s = e;
    float t = e * feat;
    #pragma unroll
    for (int off = 16; off > 0; off >>= 1) {
        s += __shfl_xor(s, off, 32);
        t += __shfl_xor(t, off, 32);
    }
    if (lane == 0) out[point] = t / s;
}

extern "C" void kernel_launch(void* const* d_in, const int* in_sizes, int n_in,
                              void* d_out, int out_size, void* d_ws, size_t ws_size,
                              hipStream_t stream) {
    const float* tp    = (const float*)d_in[0];   // (B, 56, 256, 768) f32
    const float* coord = (const float*)d_in[1];   // (B*8192*24, 3)    f32
    const int*   index = (const int*)d_in[2];     // (B, 8192)         i32
    float* out = (float*)d_out;                   // (B, 8192)         f32
    float* mf  = (float*)d_ws;                    // 786432 floats = 3 MiB scratch

    const int npix     = NB * 3 * HH * HH;        // 786432 pixels, 16 per wave
    const int nblocks1 = npix / (16 * 8);         // 6144 blocks of 8 waves
    mean_kernel<<<nblocks1, 256, 0, stream>>>(tp, mf);

    const int npoints  = NB * NPTS;               // 32768 points, 1 wave each
    const int nblocks2 = npoints / 8;             // 4096 blocks of 8 waves
    sample_kernel<<<nblocks2, 256, 0, stream>>>(tp, coord, index, mf, out);
}